// central_PairConLoss_68152541053489
// MI455X (gfx1250) — compile-verified
//
#include <hip/hip_runtime.h>

typedef float v2f __attribute__((ext_vector_type(2)));
typedef float v8f __attribute__((ext_vector_type(8)));

#define N_TOTAL 262144
#define D_DIM   128
#define C_DIM   64
#define INV_T   20.0f   // 1 / 0.05

// ---------------- kernel 0: zero counts + output scalar ----------------
__global__ void pcl_init_kernel(float* __restrict__ counts, float* __restrict__ out) {
    int i = threadIdx.x;
    if (i < C_DIM) counts[i] = 0.0f;
    if (i == 0)    out[0]    = 0.0f;
}

// ---------------- kernel 1: label histogram -> float counts ----------------
__global__ void pcl_hist_kernel(const int* __restrict__ label,
                                float* __restrict__ counts, int n) {
    __shared__ int h[C_DIM];
    int t = threadIdx.x;
    if (t < C_DIM) h[t] = 0;
    __syncthreads();
    for (int i = blockIdx.x * blockDim.x + t; i < n; i += gridDim.x * blockDim.x) {
        atomicAdd(&h[label[i] & (C_DIM - 1)], 1);
    }
    __syncthreads();
    if (t < C_DIM) {
        int c = h[t];
        if (c) atomicAdd(&counts[t], (float)c);
    }
}

// ---------------- kernel 2: fused GEMM (fp32 WMMA) + loss ----------------
// Block: 256 threads = 8 wave32 waves. Each wave owns a 16-row tile of f1.
// Per wave: S[16x64] = f1_tile[16x128] @ centralsT[128x64] via
// V_WMMA_F32_16X16X4_F32 (4 column tiles x 32 K-steps), then a fully
// branchless fused epilogue (exp/log on the TRANS pipe, co-executes w/ VALU).
__launch_bounds__(256)
__global__ void pcl_loss_kernel(const float* __restrict__ f1,
                                const float* __restrict__ centrals,
                                const int*   __restrict__ label,
                                const float* __restrict__ counts,
                                float* __restrict__ out) {
    __shared__ float ldsC[C_DIM * D_DIM];   // 32 KB: centrals, row-major

    const int tid = threadIdx.x;

    // Cooperative, coalesced stage of centrals into LDS (8 x b128 per thread).
    {
        const float4* src = (const float4*)centrals;
        float4*       dst = (float4*)ldsC;
        #pragma unroll
        for (int i = tid; i < (C_DIM * D_DIM) / 4; i += 256) dst[i] = src[i];
    }
    __syncthreads();

    const int wave = tid >> 5;
    const int lane = tid & 31;
    const int ln   = lane & 15;      // column-within-tile / row-of-A index
    const int hi   = lane >> 4;      // half-wave select (K phase / row half)

    const int r0 = (blockIdx.x * 8 + wave) * 16;     // first row of this wave's tile

    // A fragment source: lane (ln,hi) reads f1[r0+ln][4j + 2*hi .. +1] each step
    const float* arow = f1 + (size_t)(r0 + ln) * D_DIM + 2 * hi;
    // B fragment source: lane (ln,hi) reads centrals[16t+ln][4j + 2*hi .. +1]
    const float* brow = ldsC + (size_t)ln * D_DIM + 2 * hi;

    v8f acc0 = {}, acc1 = {}, acc2 = {}, acc3 = {};

    #pragma unroll 4
    for (int j = 0; j < D_DIM / 4; ++j) {
        v2f a  = *(const v2f*)(arow + 4 * j);                    // global b64
        v2f b0 = *(const v2f*)(brow + 0 * 16 * D_DIM + 4 * j);   // ds b64
        v2f b1 = *(const v2f*)(brow + 1 * 16 * D_DIM + 4 * j);
        v2f b2 = *(const v2f*)(brow + 2 * 16 * D_DIM + 4 * j);
        v2f b3 = *(const v2f*)(brow + 3 * 16 * D_DIM + 4 * j);
        acc0 = __builtin_amdgcn_wmma_f32_16x16x4_f32(false, a, false, b0, (short)0, acc0, false, false);
        acc1 = __builtin_amdgcn_wmma_f32_16x16x4_f32(false, a, false, b1, (short)0, acc1, false, false);
        acc2 = __builtin_amdgcn_wmma_f32_16x16x4_f32(false, a, false, b2, (short)0, acc2, false, false);
        acc3 = __builtin_amdgcn_wmma_f32_16x16x4_f32(false, a, false, b3, (short)0, acc3, false, false);
    }

    // ---- fused, branchless epilogue ----
    // C/D layout: acc[t][v] at lane = (hi<<4)|ln  holds  S[row = v + 8*hi][col = 16*t + ln].
    const float cn0 = counts[ln +  0];
    const float cn1 = counts[ln + 16];
    const float cn2 = counts[ln + 32];
    const float cn3 = counts[ln + 48];
    const float p0  = (cn0 > 0.0f) ? 1.0f : 0.0f;   // present masks for this lane's 4 columns
    const float p1  = (cn1 > 0.0f) ? 1.0f : 0.0f;
    const float p2  = (cn2 > 0.0f) ? 1.0f : 0.0f;
    const float p3  = (cn3 > 0.0f) ? 1.0f : 0.0f;

    float partial = 0.0f;
    const int rbase = r0 + 8 * hi;

    #pragma unroll
    for (int v = 0; v < 8; ++v) {
        const int   lbl  = label[rbase + v];        // uniform within each half-wave
        const float nl   = counts[lbl];
        const int   tsel = lbl >> 4;
        const int   csel = lbl & 15;
        const int   src  = (hi << 4) | csel;        // positive holder lane (same half)

        // broadcast all four candidates, then cndmask-select (no exec branching)
        const float sp0 = __shfl(acc0[v], src, 32);
        const float sp1 = __shfl(acc1[v], src, 32);
        const float sp2 = __shfl(acc2[v], src, 32);
        const float sp3 = __shfl(acc3[v], src, 32);
        const float sp  = (tsel == 0) ? sp0 : (tsel == 1) ? sp1 : (tsel == 2) ? sp2 : sp3;

        const float invnl = 1.0f / nl;
        // mask: present[n] && n != label  (n == lbl  <=>  t == tsel && ln == csel)
        const float w0 = ((tsel == 0) && (ln == csel)) ? 0.0f : p0 * invnl;
        const float w1 = ((tsel == 1) && (ln == csel)) ? 0.0f : p1 * invnl;
        const float w2 = ((tsel == 2) && (ln == csel)) ? 0.0f : p2 * invnl;
        const float w3 = ((tsel == 3) && (ln == csel)) ? 0.0f : p3 * invnl;

        // term = -log(pos/(pos + n_l*E_n)) = log(1 + n_l * exp((s_n - s_pos)/T))
        partial += w0 * __logf(fmaf(nl, __expf((acc0[v] - sp) * INV_T), 1.0f));
        partial += w1 * __logf(fmaf(nl, __expf((acc1[v] - sp) * INV_T), 1.0f));
        partial += w2 * __logf(fmaf(nl, __expf((acc2[v] - sp) * INV_T), 1.0f));
        partial += w3 * __logf(fmaf(nl, __expf((acc3[v] - sp) * INV_T), 1.0f));
    }

    // wave32 reduction, one atomic per wave
    #pragma unroll
    for (int off = 16; off > 0; off >>= 1)
        partial += __shfl_xor(partial, off, 32);
    if (lane == 0) atomicAdd(out, partial);
}

// ---------------- launcher ----------------
extern "C" void kernel_launch(void* const* d_in, const int* in_sizes, int n_in,
                              void* d_out, int out_size, void* d_ws, size_t ws_size,
                              hipStream_t stream) {
    const float* f1       = (const float*)d_in[0];
    const float* centrals = (const float*)d_in[1];
    const int*   label    = (const int*)d_in[2];
    float*       out      = (float*)d_out;
    float*       counts   = (float*)d_ws;    // 64 floats of scratch

    pcl_init_kernel<<<1, 64, 0, stream>>>(counts, out);
    pcl_hist_kernel<<<256, 256, 0, stream>>>(label, counts, N_TOTAL);

    // N / (16 rows/wave * 8 waves/block) = 262144 / 128 = 2048 blocks
    pcl_loss_kernel<<<2048, 256, 0, stream>>>(f1, centrals, label, counts, out);
}